// BOWRegression_89610197664088
// MI455X (gfx1250) — compile-verified
//
#include <hip/hip_runtime.h>
#include <hip/hip_bf16.h>
#include <stdint.h>

#define T_TOK           200
#define B_ROWS          4096
#define V_VOCAB         100000
#define PAD_ID          1u
#define TOKSTRIDE       201           // pad row stride to stagger LDS banks
#define WAVES_PER_BLOCK 4
#define ROWS_PER_WAVE   16

typedef float v2f __attribute__((ext_vector_type(2)));
typedef float v8f __attribute__((ext_vector_type(8)));

typedef __attribute__((address_space(1))) int* gptr_i32;
typedef __attribute__((address_space(3))) int* lptr_i32;

#if defined(__has_builtin)
#if __has_builtin(__builtin_amdgcn_global_load_async_to_lds_b32) && \
    __has_builtin(__builtin_amdgcn_s_wait_asynccnt)
#define USE_ASYNC_LDS 1
#endif
#endif
#ifndef USE_ASYNC_LDS
#define USE_ASYNC_LDS 0
#endif

__global__ __launch_bounds__(WAVES_PER_BLOCK * 32)
void bow_sigmoid_kernel(const int* __restrict__ text,   // [T, B] token ids
                        const float* __restrict__ W,    // [V]
                        const float* __restrict__ bptr, // [1]
                        float* __restrict__ out)        // [B]
{
    __shared__ uint32_t toks[WAVES_PER_BLOCK][ROWS_PER_WAVE * TOKSTRIDE];

    const int lane    = threadIdx.x & 31;
    const int w       = threadIdx.x >> 5;
    const int rowBase = (blockIdx.x * WAVES_PER_BLOCK + w) * ROWS_PER_WAVE;

    // ---- Stage this wave's 16 rows x 200 tokens into LDS via async DMA --
    // idx -> (r = idx&15, p = idx>>4): 16 consecutive lanes hit 16
    // consecutive rows of one position => contiguous 64B global segments.
    for (int idx = lane; idx < ROWS_PER_WAVE * T_TOK; idx += 32) {
        const int r = idx & 15;
        const int p = idx >> 4;
        const size_t gidx = (size_t)p * B_ROWS + (size_t)(rowBase + r);
#if USE_ASYNC_LDS
        __builtin_amdgcn_global_load_async_to_lds_b32(
            (gptr_i32)(text + gidx),
            (lptr_i32)&toks[w][r * TOKSTRIDE + p],
            0, 0);
#else
        toks[w][r * TOKSTRIDE + p] = (uint32_t)text[gidx];
#endif
    }
#if USE_ASYNC_LDS
    __builtin_amdgcn_s_wait_asynccnt(0);
#endif
    __syncthreads();

    // ---- Dedup + gather: lane (l&15) owns row rlocal; lanes 0-15 take
    // K-slots {0,1}, lanes 16-31 take K-slots {2,3} (matches WMMA A layout).
    const int rlocal = lane & 15;
    const uint32_t* rowToks = &toks[w][rlocal * TOKSTRIDE];
    const int koff = (lane >> 4) << 1;   // 0 or 2

    float acc0 = 0.0f, acc1 = 0.0f;
    for (int base = 0; base < T_TOK; base += 4) {
        const int p0 = base + koff;
        const uint32_t t0 = rowToks[p0];
        const uint32_t t1 = rowToks[p0 + 1];
        // Warm the W gathers through L2 while we do the dedup scan.
        __builtin_prefetch(&W[t0], 0, 0);
        __builtin_prefetch(&W[t1], 0, 0);
        bool c0 = (t0 != PAD_ID);
        bool c1 = (t1 != PAD_ID) & (t1 != t0);
        for (int j = 0; j < p0; ++j) {         // first-occurrence (set) test
            const uint32_t tj = rowToks[j];
            c0 = c0 & (tj != t0);
            c1 = c1 & (tj != t1);
        }
        if (c0) acc0 += W[t0];
        if (c1) acc1 += W[t1];
    }

    // ---- WMMA cross-lane reduction: D = A(16x4 partials) x B(4x16 ones) --
    // A layout: lanes 0-15 {K0,K1} in 2 VGPRs, lanes 16-31 {K2,K3}. With
    // B = all-ones, D[m][n] = row-sum of row m for every n.
    v2f a;  a.x = acc0; a.y = acc1;
    v2f bb; bb.x = 1.0f; bb.y = 1.0f;
    v8f cacc = {};
    cacc = __builtin_amdgcn_wmma_f32_16x16x4_f32(
        /*neg_a=*/false, a, /*neg_b=*/false, bb,
        /*c_mod=*/(short)0, cacc, /*reuse_a=*/false, /*reuse_b=*/false);

    // C layout: lanes 0-15 VGPR j = D[M=j][N=lane]; lanes 16-31 VGPR j =
    // D[M=8+j][N=lane-16]. Branchless 8:1 mux (7x v_cndmask) instead of a
    // divergent if/else tree; only the final store is lane-masked.
    const int i = lane & 7;
    const float s01 = (i & 1) ? cacc[1] : cacc[0];
    const float s23 = (i & 1) ? cacc[3] : cacc[2];
    const float s45 = (i & 1) ? cacc[5] : cacc[4];
    const float s67 = (i & 1) ? cacc[7] : cacc[6];
    const float s03 = (i & 2) ? s23 : s01;
    const float s47 = (i & 2) ? s67 : s45;
    const float s   = (i & 4) ? s47 : s03;

    const float bias = bptr[0];
    const float z    = s + bias;
    const float sig  = 1.0f / (1.0f + __expf(-z));
    const int   row  = rowBase + ((lane >= 16) ? 8 : 0) + i;

    if ((lane & 8) == 0) {
        out[row] = sig;
    }
}

extern "C" void kernel_launch(void* const* d_in, const int* in_sizes, int n_in,
                              void* d_out, int out_size, void* d_ws, size_t ws_size,
                              hipStream_t stream) {
    (void)in_sizes; (void)n_in; (void)out_size; (void)d_ws; (void)ws_size;
    const int*   text = (const int*)d_in[0];
    const float* W    = (const float*)d_in[1];
    const float* b    = (const float*)d_in[2];
    float* out = (float*)d_out;

    dim3 grid(B_ROWS / (WAVES_PER_BLOCK * ROWS_PER_WAVE));   // 64 blocks
    dim3 block(WAVES_PER_BLOCK * 32);                        // 128 threads (4 waves)
    hipLaunchKernelGGL(bow_sigmoid_kernel, grid, block, 0, stream, text, W, b, out);
}